// TwoCompartmentLIF_30451318128864
// MI455X (gfx1250) — compile-verified
//
#include <hip/hip_runtime.h>
#include <stdint.h>

// Two-compartment LIF scan: T=50 Euler steps over N=500000 independent
// neurons. Memory-bound (36 B / ~65 FLOP per neuron-step => ~39us roofline at
// 23.3 TB/s). Strategy: register-resident state, float4 streaming, and
// CDNA5 async global->LDS double-buffered prefetch (ASYNCcnt pipeline).

typedef float v4f __attribute__((ext_vector_type(4)));

#define TSTEPS  50
#define BLOCK   256
#define NSTREAM 7

// exp(-dt/tau) decay factors (dt = 1 ms)
#define D_E  0.8187307530779818f   // tau 5
#define D_I  0.9048374180359595f   // tau 10
#define D_N  0.9900498337491681f   // tau 100
#define D_GB 0.9975031223974601f   // tau 400
#define D_CA 0.9512294245007140f   // tau 20
#define D_AD 0.9900498337491681f   // tau 100

__device__ __forceinline__ float sigm5(float v) {
  // jax.nn.sigmoid(5*(v-0.5))
  return 1.0f / (1.0f + __expf(-5.0f * (v - 0.5f)));
}

// CDNA5 async DMA: 16B per lane from global to LDS, tracked by ASYNCcnt.
__device__ __forceinline__ void async_ld16(unsigned lds_addr,
                                           const float* base,
                                           unsigned byte_off) {
  asm volatile("global_load_async_to_lds_b128 %0, %1, %2"
               :: "v"(lds_addr), "v"(byte_off), "s"(base)
               : "memory");
}

__global__ __launch_bounds__(BLOCK)
void lif2c_kernel(const float* __restrict__ inEb, const float* __restrict__ inIb,
                  const float* __restrict__ inNb, const float* __restrict__ inGB,
                  const float* __restrict__ inEa, const float* __restrict__ inIa,
                  const float* __restrict__ inNa,
                  float* __restrict__ out, int N4) {
  // double-buffered input tiles: [buf][stream][thread] = 56 KB
  __shared__ v4f abuf[2][NSTREAM][BLOCK];

  const int tid = threadIdx.x;
  const int i4  = blockIdx.x * BLOCK + tid;
  const bool active = (i4 < N4);
  const int i4c = active ? i4 : (N4 - 1);  // clamp: keep EXEC full for async ops

  // LDS byte address of this lane's slot (low 32 bits of generic ptr = LDS offset)
  const unsigned lds0 = (unsigned)(uintptr_t)(&abuf[0][0][0]);
  const unsigned slot = lds0 + (unsigned)tid * 16u;

  auto issue_all = [&](int t, int buf) {
    const unsigned off = ((unsigned)t * (unsigned)N4 + (unsigned)i4c) * 16u;
    const unsigned b = slot + (unsigned)buf * (NSTREAM * BLOCK * 16u);
    async_ld16(b + 0u * BLOCK * 16u, inEb, off);
    async_ld16(b + 1u * BLOCK * 16u, inIb, off);
    async_ld16(b + 2u * BLOCK * 16u, inNb, off);
    async_ld16(b + 3u * BLOCK * 16u, inGB, off);
    async_ld16(b + 4u * BLOCK * 16u, inEa, off);
    async_ld16(b + 5u * BLOCK * 16u, inIa, off);
    async_ld16(b + 6u * BLOCK * 16u, inNa, off);
  };

  // register-resident state, 4 neurons per lane
  float vs[4], vd[4], gEb[4], gIb[4], gNb[4], gGB[4],
        gEa[4], gIa[4], gNa[4], gCa[4], gAd[4], ref[4];
#pragma unroll
  for (int j = 0; j < 4; ++j) {
    vs[j] = 0.0f; vd[j] = 0.0f;           // E_L = 0
    gEb[j] = gIb[j] = gNb[j] = gGB[j] = 0.0f;
    gEa[j] = gIa[j] = gNa[j] = 0.0f;
    gCa[j] = gAd[j] = ref[j] = 0.0f;
  }

  issue_all(0, 0);

  v4f* out4 = (v4f*)out;

  for (int t = 0; t < TSTEPS; ++t) {
    const int cur = t & 1;
    if (t + 1 < TSTEPS) {
      issue_all(t + 1, cur ^ 1);
      // async loads complete in order: <=7 outstanding => step t's 7 landed
      asm volatile("s_wait_asynccnt 0x7" ::: "memory");
    } else {
      asm volatile("s_wait_asynccnt 0x0" ::: "memory");
    }

    // lane-private slots: no workgroup barrier needed
    v4f q0 = abuf[cur][0][tid];
    v4f q1 = abuf[cur][1][tid];
    v4f q2 = abuf[cur][2][tid];
    v4f q3 = abuf[cur][3][tid];
    v4f q4 = abuf[cur][4][tid];
    v4f q5 = abuf[cur][5][tid];
    v4f q6 = abuf[cur][6][tid];

    v4f spk, vtr;
#pragma unroll
    for (int j = 0; j < 4; ++j) {
      // decay + inject
      gEb[j] = gEb[j] * D_E  + q0[j];
      gIb[j] = gIb[j] * D_I  + q1[j];
      gNb[j] = gNb[j] * D_N  + q2[j];
      gGB[j] = gGB[j] * D_GB + q3[j];
      gEa[j] = gEa[j] * D_E  + q4[j];
      gIa[j] = gIa[j] * D_I  + q5[j];
      gNa[j] = gNa[j] * D_N  + q6[j];
      gCa[j] *= D_CA;

      float v_s = vs[j], v_d = vd[j];
      const float ms = sigm5(v_s);
      const float md = sigm5(v_d);

      // somatic current (C_m = 1)
      float Is = 0.05f * (0.0f - v_s)
               + gEb[j] * (3.0f - v_s)
               + gIb[j] * (-0.5f - v_s)
               + gNb[j] * ms * (3.0f - v_s)
               + gGB[j] * (-0.8f - v_s)
               + gAd[j] * (-0.5f - v_s)
               + 0.05f * (v_d - v_s);
      // dendritic current
      float Id = 0.03f * (0.0f - v_d)
               + gEa[j] * (3.0f - v_d)
               + gIa[j] * (-0.5f - v_d)
               + gNa[j] * md * (3.0f - v_d)
               + gCa[j] * (4.0f - v_d)
               + 0.05f * (v_s - v_d);

      const bool refr = ref[j] > 0.0f;
      v_s = refr ? 0.0f : (v_s + Is);       // DT = 1
      v_d = v_d + 2.0f * Id;                // DT / C_D = 2

      const bool spike = (v_s >= 1.0f) && (!refr);
      const float sf = spike ? 1.0f : 0.0f;
      v_s = spike ? 0.0f : v_s;             // V_RESET = 0
      ref[j] = spike ? 5.0f : fmaxf(ref[j] - 1.0f, 0.0f);
      gAd[j] = gAd[j] * D_AD;               // ADAPT_INC == 0
      v_d = spike ? (v_d + 0.3f * (4.0f - v_d)) : v_d;   // bAP
      gCa[j] += (v_d >= 2.0f) ? 0.3f : 0.0f;             // Ca spike

      vs[j] = v_s; vd[j] = v_d;
      spk[j] = sf; vtr[j] = v_s;
    }

    if (active) {
      // write-once output streams: non-temporal to spare L2
      __builtin_nontemporal_store(spk, out4 + (size_t)t * N4 + i4);
      __builtin_nontemporal_store(vtr, out4 + (size_t)(TSTEPS + t) * N4 + i4);
    }
  }
}

extern "C" void kernel_launch(void* const* d_in, const int* in_sizes, int n_in,
                              void* d_out, int out_size, void* d_ws, size_t ws_size,
                              hipStream_t stream) {
  (void)n_in; (void)out_size; (void)d_ws; (void)ws_size;
  const float* inEb = (const float*)d_in[0];
  const float* inIb = (const float*)d_in[1];
  const float* inNb = (const float*)d_in[2];
  const float* inGB = (const float*)d_in[3];
  const float* inEa = (const float*)d_in[4];
  const float* inIa = (const float*)d_in[5];
  const float* inNa = (const float*)d_in[6];

  const int N  = in_sizes[0] / TSTEPS;  // inputs are [T, N]
  const int N4 = N / 4;                 // N = 500000 is divisible by 4
  const int blocks = (N4 + BLOCK - 1) / BLOCK;

  lif2c_kernel<<<blocks, BLOCK, 0, stream>>>(
      inEb, inIb, inNb, inGB, inEa, inIa, inNa, (float*)d_out, N4);
}